// GraphConv_13013750907275
// MI455X (gfx1250) — compile-verified
//
#include <hip/hip_runtime.h>
#include <hip/hip_bf16.h>

typedef __attribute__((ext_vector_type(2))) float v2f;
typedef __attribute__((ext_vector_type(8))) float v8f;

#define NODES 100000
#define DIM   256
#define EPS_N 1e-12f

// ---------------------------------------------------------------------------
// Kernel 0: zero the accumulation buffer (d_out doubles as agg buffer)
// ---------------------------------------------------------------------------
__global__ void zero_kernel(float4* __restrict__ p, int n4) {
    int i = blockIdx.x * blockDim.x + threadIdx.x;
    int stride = gridDim.x * blockDim.x;
    float4 z = make_float4(0.f, 0.f, 0.f, 0.f);
    for (; i < n4; i += stride) p[i] = z;
}

// ---------------------------------------------------------------------------
// Kernel 1: X = inputs @ W using V_WMMA_F32_16X16X4_F32 (exact fp32)
// grid.x = NODES/16 = 6250 M-tiles; block = 512 threads = 16 waves,
// wave w computes the 16x16 C tile at column block w (DIM/16 = 16 tiles).
// ---------------------------------------------------------------------------
__global__ void __launch_bounds__(512)
gemm_wmma_f32(const float* __restrict__ A,   // [NODES, DIM] row-major
              const float* __restrict__ W,   // [DIM, DIM] row-major
              float* __restrict__ X)         // [NODES, DIM]
{
    const int wave = threadIdx.x >> 5;        // 0..15 -> N tile
    const int lane = threadIdx.x & 31;
    const int lm   = lane & 15;               // M (for A) / N (for B,C)
    const int kh   = (lane >> 4) << 1;        // K sub-offset: 0 or 2
    const int mBase = blockIdx.x << 4;
    const int nBase = wave << 4;

    const float* arow = A + (size_t)(mBase + lm) * DIM;      // A row, this lane
    const float* bcol = W + (size_t)nBase + lm;              // B col, this lane

    v8f acc = {};
    #pragma unroll 4
    for (int k0 = 0; k0 < DIM; k0 += 4) {
        v2f a, b;
        // A fragment: 16x4, lane half selects K pair -> aligned float2
        float2 av = *(const float2*)(arow + k0 + kh);
        a.x = av.x; a.y = av.y;
        // B fragment: 4x16, rows k0+kh, k0+kh+1 of W at column nBase+lm
        b.x = bcol[(size_t)(k0 + kh) * DIM];
        b.y = bcol[(size_t)(k0 + kh + 1) * DIM];
        acc = __builtin_amdgcn_wmma_f32_16x16x4_f32(
            /*neg_a=*/false, a, /*neg_b=*/false, b,
            /*c_mod=*/(short)0, acc, /*reuse_a=*/false, /*reuse_b=*/false);
    }

    // C/D layout: VGPR r -> M = r + 8*(lane>=16), N = lane&15
    const int rowHalf = (lane >> 4) << 3;
    float* xout = X + (size_t)mBase * DIM + nBase;
    #pragma unroll
    for (int r = 0; r < 8; ++r)
        xout[(size_t)(r + rowHalf) * DIM + lm] = acc[r];
}

// ---------------------------------------------------------------------------
// Kernel 2: SpMM scatter  out[row] += val * X[col]
// 64 threads per edge, 4 floats per thread (float4 gather + 4 f32 atomics).
// Output buffer (102 MB) fits in the 192 MB L2, so atomics resolve in L2.
// ---------------------------------------------------------------------------
__global__ void __launch_bounds__(256)
spmm_scatter(const int* __restrict__ rows,
             const int* __restrict__ cols,
             const float* __restrict__ vals,
             const float* __restrict__ X,
             float* __restrict__ out, int E)
{
    int tid = blockIdx.x * blockDim.x + threadIdx.x;
    int e = tid >> 6;
    if (e >= E) return;
    int t = (tid & 63) << 2;                 // 0,4,...,252

    int   r = rows[e];
    int   c = cols[e];
    float v = vals[e];

    float4 xv = *(const float4*)(X + (size_t)c * DIM + t);
    float* o = out + (size_t)r * DIM + t;
    atomicAdd(o + 0, v * xv.x);
    atomicAdd(o + 1, v * xv.y);
    atomicAdd(o + 2, v * xv.z);
    atomicAdd(o + 3, v * xv.w);
}

// ---------------------------------------------------------------------------
// Kernel 3: h = relu(agg + b); out = h / max(||h||_2, eps)   (in place)
// One wave32 per row, 8 floats per lane, shfl_xor tree reduction.
// ---------------------------------------------------------------------------
__global__ void __launch_bounds__(256)
finalize_norm(float* __restrict__ out, const float* __restrict__ bias, int n)
{
    const int wave = threadIdx.x >> 5;
    const int lane = threadIdx.x & 31;
    const int row  = (blockIdx.x << 3) + wave;
    if (row >= n) return;

    float* o = out + (size_t)row * DIM;
    const int off = lane << 3;               // 8 floats per lane

    float4 h0 = *(const float4*)(o + off);
    float4 h1 = *(const float4*)(o + off + 4);
    float4 b0 = *(const float4*)(bias + off);
    float4 b1 = *(const float4*)(bias + off + 4);

    h0.x = fmaxf(h0.x + b0.x, 0.f); h0.y = fmaxf(h0.y + b0.y, 0.f);
    h0.z = fmaxf(h0.z + b0.z, 0.f); h0.w = fmaxf(h0.w + b0.w, 0.f);
    h1.x = fmaxf(h1.x + b1.x, 0.f); h1.y = fmaxf(h1.y + b1.y, 0.f);
    h1.z = fmaxf(h1.z + b1.z, 0.f); h1.w = fmaxf(h1.w + b1.w, 0.f);

    float s = h0.x*h0.x + h0.y*h0.y + h0.z*h0.z + h0.w*h0.w
            + h1.x*h1.x + h1.y*h1.y + h1.z*h1.z + h1.w*h1.w;
    #pragma unroll
    for (int d = 16; d > 0; d >>= 1)
        s += __shfl_xor(s, d, 32);

    float scale = 1.0f / fmaxf(sqrtf(s), EPS_N);
    h0.x *= scale; h0.y *= scale; h0.z *= scale; h0.w *= scale;
    h1.x *= scale; h1.y *= scale; h1.z *= scale; h1.w *= scale;
    *(float4*)(o + off)     = h0;
    *(float4*)(o + off + 4) = h1;
}

// ---------------------------------------------------------------------------
extern "C" void kernel_launch(void* const* d_in, const int* in_sizes, int n_in,
                              void* d_out, int out_size, void* d_ws, size_t ws_size,
                              hipStream_t stream) {
    const float* inputs = (const float*)d_in[0];   // [N, D]
    const int*   arows  = (const int*)  d_in[1];   // [E]
    const int*   acols  = (const int*)  d_in[2];   // [E]
    const float* avals  = (const float*)d_in[3];   // [E]
    const float* W      = (const float*)d_in[4];   // [D, D]
    const float* bias   = (const float*)d_in[5];   // [D]
    float*       out    = (float*)d_out;           // [N, D]
    float*       X      = (float*)d_ws;            // [N, D] scratch for inputs@W

    const int E = in_sizes[1];

    // 0) out = 0 (it is the SpMM accumulator)
    const int n4 = NODES * DIM / 4;
    zero_kernel<<<2048, 256, 0, stream>>>((float4*)out, n4);

    // 1) X = inputs @ W   (fp32 WMMA)
    gemm_wmma_f32<<<NODES / 16, 512, 0, stream>>>(inputs, W, X);

    // 2) out[r] += val * X[c]  per edge (64 threads/edge)
    {
        long long total = (long long)E * 64;
        int blocks = (int)((total + 255) / 256);
        spmm_scatter<<<blocks, 256, 0, stream>>>(arows, acols, avals, X, out, E);
    }

    // 3) relu(out + b), row-wise L2 normalize, in place
    finalize_norm<<<(NODES + 7) / 8, 256, 0, stream>>>(out, bias, NODES);
}